// prj_module00_90589450207262
// MI455X (gfx1250) — compile-verified
//
#include <hip/hip_runtime.h>
#include <stdint.h>

// Fan-beam CT forward projector for MI455X (gfx1250).
// Strategy: stage the whole 256x256 f32 image (256 KB) into LDS (320 KB/WGP on
// CDNA5) using gfx1250 async global->LDS copies (ASYNCcnt-tracked), then do all
// bilinear gathers from LDS. One block per (view, batch), one thread per
// detector. Inner sample loop unrolled 4x to keep ~16 independent ds_load_b32
// in flight per wave (only 4 waves/SIMD fit due to the 256 KB LDS footprint).

#define VIEWS 512
#define DETS  512
#define IMG_H 256
#define IMG_W 256
#define NS    384
#define TPB   512   // threads per block == DETS (16 wave32s)

__device__ __forceinline__ float sample_lds(const float* s, int y, int x) {
    // Reference semantics: value = im[clip(y), clip(x)] * inside
    const bool inside = ((unsigned)x < (unsigned)IMG_W) && ((unsigned)y < (unsigned)IMG_H);
    int xc = x < 0 ? 0 : (x > IMG_W - 1 ? IMG_W - 1 : x);
    int yc = y < 0 ? 0 : (y > IMG_H - 1 ? IMG_H - 1 : y);
    float v = s[yc * IMG_W + xc];   // lowers to ds_load_b32 (always in-range)
    return inside ? v : 0.0f;
}

__device__ __forceinline__ float bilinear_tap(const float* smem, float px, float py) {
    const float fx = floorf(px);
    const float fy = floorf(py);
    const float wx = px - fx;
    const float wy = py - fy;
    const int   xi = (int)fx;
    const int   yi = (int)fy;

    const float v00 = sample_lds(smem, yi,     xi);
    const float v01 = sample_lds(smem, yi,     xi + 1);
    const float v10 = sample_lds(smem, yi + 1, xi);
    const float v11 = sample_lds(smem, yi + 1, xi + 1);

    // Exact reference weighting expression
    return v00 * (1.0f - wx) * (1.0f - wy)
         + v01 * wx          * (1.0f - wy)
         + v10 * (1.0f - wx) * wy
         + v11 * wx          * wy;
}

__global__ __launch_bounds__(TPB) void fanbeam_project_kernel(
    const float* __restrict__ img,   // (2,1,256,256) f32
    const float* __restrict__ opt,   // 12 f32 options
    float* __restrict__ out)         // (2,1,512,512) f32
{
    extern __shared__ float smem[];  // IMG_H*IMG_W floats = 256 KB

    const int tid  = threadIdx.x;    // detector index
    const int view = blockIdx.x;     // view index
    const int b    = blockIdx.y;     // batch index

    const float* gimg = img + (size_t)b * (IMG_H * IMG_W);

    // ---- Stage image into LDS via CDNA5 async global->LDS copies ----------
    // Flat pointer to LDS: low 32 bits are the wave-relative LDS byte offset
    // (aperture mapping truncates), which is what the async op's VDST expects.
    const uint32_t lds_base = (uint32_t)(uintptr_t)smem;
    #pragma unroll
    for (int i = 0; i < (IMG_H * IMG_W * 4) / (TPB * 16); ++i) {   // 32 iters
        const uint32_t off      = (uint32_t)((i * TPB + tid) * 16);
        const uint32_t lds_addr = lds_base + off;
        const uint64_t gaddr    = (uint64_t)(uintptr_t)((const char*)gimg + off);
        asm volatile("global_load_async_to_lds_b128 %0, %1, off"
                     :: "v"(lds_addr), "v"(gaddr)
                     : "memory");
    }
    asm volatile("s_wait_asynccnt 0x0" ::: "memory");
    __syncthreads();

    // ---- Ray geometry (matches reference, f32 throughout) ------------------
    const float dImg     = opt[4];
    const float dDet     = opt[5];
    const float ang0     = opt[6];
    const float dAng     = opt[7];
    const float s2r      = opt[8];
    const float d2r      = opt[9];
    const float binshift = opt[10];

    const float beta = ang0 + dAng * (float)view;
    float sb, cb;
    sincosf(beta, &sb, &cb);

    const float u    = ((float)tid - (float)(DETS - 1) * 0.5f) * dDet + binshift;
    const float srcx = s2r * cb;
    const float srcy = s2r * sb;
    const float detx = -d2r * cb - u * sb;
    const float dety = -d2r * sb + u * cb;

    float dx = detx - srcx;
    float dy = dety - srcy;
    const float invn = 1.0f / sqrtf(dx * dx + dy * dy);
    dx *= invn;
    dy *= invn;

    const float R    = 0.5f * sqrtf((float)(IMG_H * IMG_H + IMG_W * IMG_W)) * dImg;
    const float dl   = 2.0f * R / (float)NS;
    const float t0   = s2r - R;
    const float invD = 1.0f / dImg;
    const float cx   = (float)(IMG_W - 1) * 0.5f;
    const float cy   = (float)(IMG_H - 1) * 0.5f;

    float acc = 0.0f;
    #pragma unroll 4
    for (int s = 0; s < NS; ++s) {
        const float t  = t0 + ((float)s + 0.5f) * dl;
        const float px = fmaf(fmaf(dx, t, srcx), invD, cx);
        const float py = fmaf(fmaf(dy, t, srcy), invD, cy);
        acc += bilinear_tap(smem, px, py);
    }

    // Output layout (2,1,VIEWS,DETS)
    out[((size_t)b * VIEWS + (size_t)view) * DETS + (size_t)tid] = acc * dl;
}

extern "C" void kernel_launch(void* const* d_in, const int* in_sizes, int n_in,
                              void* d_out, int out_size, void* d_ws, size_t ws_size,
                              hipStream_t stream) {
    (void)in_sizes; (void)n_in; (void)d_ws; (void)ws_size; (void)out_size;
    const float* img = (const float*)d_in[0];   // (2,1,256,256) f32
    const float* opt = (const float*)d_in[1];   // 12 f32
    float* out = (float*)d_out;                 // (2,1,512,512) f32

    const size_t shmem = (size_t)IMG_H * IMG_W * sizeof(float);   // 256 KB

    // Opt in to >64KB dynamic LDS (320 KB per WGP on CDNA5); ignore error if
    // the runtime doesn't require it.
    (void)hipFuncSetAttribute((const void*)fanbeam_project_kernel,
                              hipFuncAttributeMaxDynamicSharedMemorySize,
                              (int)shmem);

    dim3 grid(VIEWS, 2, 1);
    dim3 block(TPB, 1, 1);
    fanbeam_project_kernel<<<grid, block, shmem, stream>>>(img, opt, out);
}